// Model_38062000177879
// MI455X (gfx1250) — compile-verified
//
#include <hip/hip_runtime.h>
#include <math.h>

#define HIDN 150     // hidden_size
#define PADH 152     // HIDN padded to multiple of 4 (K for WMMA)
#define HGRU 256     // GRU hidden
#define BN   16      // batch
#define LN   32      // seq len
#define LP   33      // prefix length (L+1)
#define DMODEL 512

typedef float v2f __attribute__((ext_vector_type(2)));
typedef float v8f __attribute__((ext_vector_type(8)));

struct GruP { const float* wih; const float* whh; const float* bih; const float* bhh; };

__device__ __forceinline__ float sigf(float x) { return 1.0f / (1.0f + expf(-x)); }

__device__ __forceinline__ v8f vzero8() {
    v8f z = {0.f,0.f,0.f,0.f,0.f,0.f,0.f,0.f};
    return z;
}

// ---------------------------------------------------------------------------
// f32 WMMA core (V_WMMA_F32_16X16X4_F32), K a compile-time multiple of 4.
// A[16,K] row-major (lda), W[16 rows, K] row-major (ldw); computes A @ W^T.
// ISA layout: A lane t: M=t&15, K=k0+(t>>4)*2+{0,1}; B lane t: N=t&15, same K.
// D: vgpr r -> M = r + (t>>4)*8, N = t&15.  No guards: rows pre-clamped,
// pads pre-zeroed, fragments are unconditional 64-bit loads.
// ---------------------------------------------------------------------------
template<int K>
__device__ __forceinline__ v8f wmma_mac(const float* A, int lda,
                                        const float* W, int ldw, v8f acc)
{
    const int lane = threadIdx.x & 31;
    const int rsel = lane & 15;
    const int kb   = (lane >> 4) << 1;
    const float* arow = A + (size_t)rsel * lda;
    const float* wrow = W + (size_t)rsel * ldw;
    #pragma unroll 4
    for (int k0 = 0; k0 < K; k0 += 4) {
        v2f a = *(const v2f*)(arow + k0 + kb);
        v2f b = *(const v2f*)(wrow + k0 + kb);
        acc = __builtin_amdgcn_wmma_f32_16x16x4_f32(false, a, false, b,
                                                    (short)0, acc, false, false);
    }
    return acc;
}

__device__ __forceinline__ void wmma_store_f32(float* C, int ldc, v8f d,
                                               const float* bias,
                                               int mvalid, int nvalid)
{
    const int lane = threadIdx.x & 31;
    const int n    = lane & 15;
    const int mo   = (lane >> 4) << 3;
    float bv = 0.0f;
    if (bias && n < nvalid) bv = bias[n];
    #pragma unroll
    for (int r = 0; r < 8; ++r) {
        int m = r + mo;
        if (m < mvalid && n < nvalid)
            C[(size_t)m * ldc + n] = d[r] + bv;
    }
}

// ---------------------------------------------------------------------------
// GEMM: C[M, N] = A[M, K] @ W[N, K]^T + bias;  C row stride = ldc.
// Each wave: one 16-row M tile x four 16-col N tiles (A fragment reused 4x).
// ---------------------------------------------------------------------------
template<int K>
__global__ void gemm_wmma4_kernel(const float* __restrict__ A,
                                  const float* __restrict__ W,
                                  const float* __restrict__ bias,
                                  float* __restrict__ C,
                                  int M, int N, int ldc)
{
    const int wpb    = blockDim.x >> 5;
    const int wave   = blockIdx.x * wpb + (threadIdx.x >> 5);
    const int nwaves = gridDim.x * wpb;
    const int Mt = (M + 15) >> 4;
    const int Ng = (N + 63) >> 6;          // groups of 4 n-tiles
    const int lane = threadIdx.x & 31;
    const int rsel = lane & 15;
    const int kb   = (lane >> 4) << 1;

    for (int job = wave; job < Mt * Ng; job += nwaves) {
        const int mt = job / Ng, ng = job - mt * Ng;
        const int m0 = mt << 4, n0 = ng << 6;

        int mrow = m0 + rsel; if (mrow >= M) mrow = M - 1;
        const float* arow = A + (size_t)mrow * K;
        const float* wr[4];
        #pragma unroll
        for (int j = 0; j < 4; ++j) {
            int nr = n0 + j * 16 + rsel; if (nr >= N) nr = N - 1;
            wr[j] = W + (size_t)nr * K;
        }

        v8f acc[4];
        #pragma unroll
        for (int j = 0; j < 4; ++j) acc[j] = vzero8();

        #pragma unroll 2
        for (int k0 = 0; k0 < K; k0 += 4) {
            v2f a = *(const v2f*)(arow + k0 + kb);
            #pragma unroll
            for (int j = 0; j < 4; ++j) {
                v2f b = *(const v2f*)(wr[j] + k0 + kb);
                acc[j] = __builtin_amdgcn_wmma_f32_16x16x4_f32(false, a, false, b,
                                                               (short)0, acc[j],
                                                               false, false);
            }
        }

        const int mv = (M - m0 < 16) ? (M - m0) : 16;
        #pragma unroll
        for (int j = 0; j < 4; ++j) {
            int nn0 = n0 + j * 16;
            int nv = N - nn0; if (nv > 16) nv = 16;
            if (nv > 0)
                wmma_store_f32(C + (size_t)m0 * ldc + nn0, ldc, acc[j],
                               bias ? bias + nn0 : nullptr, mv, nv);
        }
    }
}

// ---------------------------------------------------------------------------
// Repack weight [rows, 150] -> [rows, 152], pad columns zeroed.
// ---------------------------------------------------------------------------
__global__ void repack_kernel(const float* __restrict__ src,
                              float* __restrict__ dst, int rows)
{
    int i = blockIdx.x * blockDim.x + threadIdx.x;
    int total = rows * PADH;
    int nthr = gridDim.x * blockDim.x;
    for (int k = i; k < total; k += nthr) {
        int r = k / PADH, c = k - r * PADH;
        dst[k] = (c < HIDN) ? src[r * HIDN + c] : 0.0f;
    }
}

// ---------------------------------------------------------------------------
// Embedding gather: dt[L, B, PADH] = emb[x[b, t]], pads zero.
// ---------------------------------------------------------------------------
__global__ void embed_kernel(const int* __restrict__ x,
                             const float* __restrict__ emb,
                             float* __restrict__ dt)
{
    int i = blockIdx.x * blockDim.x + threadIdx.x;
    int total = LN * BN * PADH;
    if (i >= total) return;
    int d = i % PADH;
    int b = (i / PADH) % BN;
    int t = i / (PADH * BN);
    int tok = x[b * LN + t];
    dt[i] = (d < HIDN) ? emb[(size_t)tok * HIDN + d] : 0.0f;
}

// ---------------------------------------------------------------------------
// Encoder GRU layer (both directions per launch; blockIdx.x = dir).
// xseq: [L, B, IND].  yout (optional): [L, B, 512].  endt: [4, B, 256].
// ---------------------------------------------------------------------------
template<int IND>
__global__ void enc_gru_kernel(const float* __restrict__ xseq,
                               GruP pf, GruP pb,
                               float* __restrict__ yout,
                               float* __restrict__ endt, int row_f, int row_b)
{
    __shared__ float sh_h[BN][HGRU];
    __shared__ float sh_gsum[BN][DMODEL];
    __shared__ float sh_inn[BN][HGRU];
    __shared__ float sh_hn[BN][HGRU];

    const int dir = blockIdx.x;
    GruP P = dir ? pb : pf;
    const int ycol = dir ? HGRU : 0;

    for (int i = threadIdx.x; i < BN * HGRU; i += blockDim.x)
        (&sh_h[0][0])[i] = 0.0f;
    __syncthreads();

    const int wave = threadIdx.x >> 5;

    for (int s = 0; s < LN; ++s) {
        const int t = dir ? (LN - 1 - s) : s;
        const float* xt = xseq + (size_t)t * BN * IND;

        // 64 tile jobs: 0..31 gsum (r,z), 32..47 inn, 48..63 hn
        for (int job = wave; job < 64; job += 8) {
            if (job < 32) {
                int n0 = job << 4;
                v8f acc = vzero8();
                acc = wmma_mac<IND>(xt, IND, P.wih + (size_t)n0 * IND, IND, acc);
                acc = wmma_mac<HGRU>(&sh_h[0][0], HGRU,
                                     P.whh + (size_t)n0 * HGRU, HGRU, acc);
                wmma_store_f32(&sh_gsum[0][0] + n0, DMODEL, acc, nullptr, 16, 16);
            } else if (job < 48) {
                int n0 = (job - 32) << 4;
                v8f acc = vzero8();
                acc = wmma_mac<IND>(xt, IND,
                                    P.wih + (size_t)(DMODEL + n0) * IND, IND, acc);
                wmma_store_f32(&sh_inn[0][0] + n0, HGRU, acc, nullptr, 16, 16);
            } else {
                int n0 = (job - 48) << 4;
                v8f acc = vzero8();
                acc = wmma_mac<HGRU>(&sh_h[0][0], HGRU,
                                     P.whh + (size_t)(DMODEL + n0) * HGRU, HGRU, acc);
                wmma_store_f32(&sh_hn[0][0] + n0, HGRU, acc, nullptr, 16, 16);
            }
        }
        __syncthreads();

        for (int i = threadIdx.x; i < BN * HGRU; i += blockDim.x) {
            int b = i >> 8;
            int j = i & 255;
            float r  = sigf(sh_gsum[b][j]        + P.bih[j]        + P.bhh[j]);
            float z  = sigf(sh_gsum[b][HGRU + j] + P.bih[HGRU + j] + P.bhh[HGRU + j]);
            float nn = tanhf(sh_inn[b][j] + P.bih[2 * HGRU + j] +
                             r * (sh_hn[b][j] + P.bhh[2 * HGRU + j]));
            float hnew = (1.0f - z) * nn + z * sh_h[b][j];
            sh_h[b][j] = hnew;
            if (yout)
                yout[((size_t)t * BN + b) * DMODEL + ycol + j] = hnew;
        }
        __syncthreads();
    }

    const int erow = dir ? row_b : row_f;
    for (int i = threadIdx.x; i < BN * HGRU; i += blockDim.x) {
        int b = i >> 8, j = i & 255;
        endt[((size_t)erow * BN + b) * HGRU + j] = sh_h[b][j];
    }
}

// ---------------------------------------------------------------------------
// Conv1d(4,2,k=1) decoder-hidden init + zero itinfo/data (incl. pads) + avail.
// ---------------------------------------------------------------------------
__global__ void conv_init_kernel(const float* __restrict__ endt,
                                 const float* __restrict__ conv_w,
                                 const float* __restrict__ conv_b,
                                 float* __restrict__ ha, float* __restrict__ hb,
                                 float* __restrict__ itinfo,
                                 float* __restrict__ data,
                                 int* __restrict__ avail)
{
    int i = blockIdx.x * blockDim.x + threadIdx.x;
    int nthr = gridDim.x * blockDim.x;
    if (i < BN * HGRU) {
        int b = i >> 8, j = i & 255;
        float s0 = conv_b[0], s1 = conv_b[1];
        for (int c = 0; c < 4; ++c) {
            float e = endt[((size_t)c * BN + b) * HGRU + j];
            s0 += conv_w[c]     * e;
            s1 += conv_w[4 + c] * e;
        }
        ha[i] = s0; hb[i] = s1;
    }
    for (int k = i; k < BN * LP * PADH; k += nthr) { itinfo[k] = 0.0f; data[k] = 0.0f; }
    if (i < BN * LN) avail[i] = 1;
}

// ---------------------------------------------------------------------------
// Attention (faithful Softmax(dim=1) over queries): per-batch block.
// qq/kk/vv/data: [B, LP, PADH]; writes data rows q <= idx (cols < 150).
// ---------------------------------------------------------------------------
__global__ void attn_kernel(const float* __restrict__ qq,
                            const float* __restrict__ kk,
                            const float* __restrict__ vv,
                            float* __restrict__ dataout, int idx)
{
    __shared__ float gx[LP][LP + 1];
    const int b = blockIdx.x;
    const float* Q  = qq + (size_t)b * LP * PADH;
    const float* Km = kk + (size_t)b * LP * PADH;
    const float* V  = vv + (size_t)b * LP * PADH;
    const int nv = idx + 1;

    for (int e = threadIdx.x; e < LP * LP; e += blockDim.x) {
        int q = e / LP, k = e - q * LP;
        float s = 0.0f;
        if (q < nv)
            for (int d = 0; d < HIDN; ++d) s += Q[q * PADH + d] * Km[k * PADH + d];
        gx[q][k] = s;
    }
    __syncthreads();

    for (int k = threadIdx.x; k < LP; k += blockDim.x) {
        float mx = -3.0e38f;
        for (int q = 0; q < nv; ++q) mx = fmaxf(mx, gx[q][k]);
        float sum = 0.0f;
        for (int q = 0; q < nv; ++q) { float e = expf(gx[q][k] - mx); gx[q][k] = e; sum += e; }
        float inv = (k < nv) ? (1.0f / sum) : 0.0f;   // also drops padded key columns
        for (int q = 0; q < nv; ++q) gx[q][k] *= inv;
        for (int q = nv; q < LP; ++q) gx[q][k] = 0.0f;
    }
    __syncthreads();

    for (int e = threadIdx.x; e < nv * HIDN; e += blockDim.x) {
        int q = e / HIDN, d = e - q * HIDN;
        float s = 0.0f;
        for (int k = 0; k < nv; ++k) s += gx[q][k] * V[k * PADH + d];
        dataout[((size_t)b * LP + q) * PADH + d] = s;
    }
}

// ---------------------------------------------------------------------------
// Decoder 2-layer GRU scan over prefix steps t = 0..idx.  Single block;
// hidden state carried in LDS, persisted to global between outer steps.
// dec0 wih is the 152-padded repack.
// ---------------------------------------------------------------------------
__global__ void dec_step_kernel(const float* __restrict__ data, int idx,
                                GruP p0, GruP p1,
                                float* __restrict__ ha_g, float* __restrict__ hb_g)
{
    __shared__ float sha[BN][HGRU];
    __shared__ float shb[BN][HGRU];
    __shared__ float sna[BN][HGRU];
    __shared__ float gsum[BN][DMODEL];
    __shared__ float sinn[BN][HGRU];
    __shared__ float shn[BN][HGRU];

    for (int i = threadIdx.x; i < BN * HGRU; i += blockDim.x) {
        (&sha[0][0])[i] = ha_g[i];
        (&shb[0][0])[i] = hb_g[i];
    }
    __syncthreads();

    const int wave = threadIdx.x >> 5;

    for (int t = 0; t <= idx; ++t) {
        // ---- cell 0: x = data[:, t, :] (K=152), h = sha ----
        const float* xt = data + (size_t)t * PADH;   // row b at b*LP*PADH
        for (int job = wave; job < 64; job += 8) {
            if (job < 32) {
                int n0 = job << 4;
                v8f acc = vzero8();
                acc = wmma_mac<PADH>(xt, LP * PADH,
                                     p0.wih + (size_t)n0 * PADH, PADH, acc);
                acc = wmma_mac<HGRU>(&sha[0][0], HGRU,
                                     p0.whh + (size_t)n0 * HGRU, HGRU, acc);
                wmma_store_f32(&gsum[0][0] + n0, DMODEL, acc, nullptr, 16, 16);
            } else if (job < 48) {
                int n0 = (job - 32) << 4;
                v8f acc = vzero8();
                acc = wmma_mac<PADH>(xt, LP * PADH,
                                     p0.wih + (size_t)(DMODEL + n0) * PADH, PADH, acc);
                wmma_store_f32(&sinn[0][0] + n0, HGRU, acc, nullptr, 16, 16);
            } else {
                int n0 = (job - 48) << 4;
                v8f acc = vzero8();
                acc = wmma_mac<HGRU>(&sha[0][0], HGRU,
                                     p0.whh + (size_t)(DMODEL + n0) * HGRU, HGRU, acc);
                wmma_store_f32(&shn[0][0] + n0, HGRU, acc, nullptr, 16, 16);
            }
        }
        __syncthreads();
        for (int i = threadIdx.x; i < BN * HGRU; i += blockDim.x) {
            int b = i >> 8, j = i & 255;
            float r  = sigf(gsum[b][j]        + p0.bih[j]        + p0.bhh[j]);
            float z  = sigf(gsum[b][HGRU + j] + p0.bih[HGRU + j] + p0.bhh[HGRU + j]);
            float nn = tanhf(sinn[b][j] + p0.bih[2 * HGRU + j] +
                             r * (shn[b][j] + p0.bhh[2 * HGRU + j]));
            sna[b][j] = (1.0f - z) * nn + z * sha[b][j];
        }
        __syncthreads();

        // ---- cell 1: x = sna (K=256), h = shb ----
        for (int job = wave; job < 64; job += 8) {
            if (job < 32) {
                int n0 = job << 4;
                v8f acc = vzero8();
                acc = wmma_mac<HGRU>(&sna[0][0], HGRU,
                                     p1.wih + (size_t)n0 * HGRU, HGRU, acc);
                acc = wmma_mac<HGRU>(&shb[0][0], HGRU,
                                     p1.whh + (size_t)n0 * HGRU, HGRU, acc);
                wmma_store_f32(&gsum[0][0] + n0, DMODEL, acc, nullptr, 16, 16);
            } else if (job < 48) {
                int n0 = (job - 32) << 4;
                v8f acc = vzero8();
                acc = wmma_mac<HGRU>(&sna[0][0], HGRU,
                                     p1.wih + (size_t)(DMODEL + n0) * HGRU, HGRU, acc);
                wmma_store_f32(&sinn[0][0] + n0, HGRU, acc, nullptr, 16, 16);
            } else {
                int n0 = (job - 48) << 4;
                v8f acc = vzero8();
                acc = wmma_mac<HGRU>(&shb[0][0], HGRU,
                                     p1.whh + (size_t)(DMODEL + n0) * HGRU, HGRU, acc);
                wmma_store_f32(&shn[0][0] + n0, HGRU, acc, nullptr, 16, 16);
            }
        }
        __syncthreads();
        for (int i = threadIdx.x; i < BN * HGRU; i += blockDim.x) {
            int b = i >> 8, j = i & 255;
            float r  = sigf(gsum[b][j]        + p1.bih[j]        + p1.bhh[j]);
            float z  = sigf(gsum[b][HGRU + j] + p1.bih[HGRU + j] + p1.bhh[HGRU + j]);
            float nn = tanhf(sinn[b][j] + p1.bih[2 * HGRU + j] +
                             r * (shn[b][j] + p1.bhh[2 * HGRU + j]));
            shb[b][j] = (1.0f - z) * nn + z * shb[b][j];
            sha[b][j] = sna[b][j];
        }
        __syncthreads();
    }

    for (int i = threadIdx.x; i < BN * HGRU; i += blockDim.x) {
        ha_g[i] = (&sha[0][0])[i];
        hb_g[i] = (&shb[0][0])[i];
    }
}

// ---------------------------------------------------------------------------
// Pointer head: gather z_w rows at input token ids (only <=32 of the 50257
// vocab columns are ever consumed), masked first-argmax, update state.
// One wave32 per batch element.
// ---------------------------------------------------------------------------
__global__ void score_kernel(const float* __restrict__ ha,
                             const float* __restrict__ hb,
                             const float* __restrict__ z_w,
                             const float* __restrict__ z_b,
                             const int* __restrict__ x,
                             const float* __restrict__ emb,
                             int* __restrict__ avail,
                             float* __restrict__ itinfo,
                             float* __restrict__ out, int idx)
{
    __shared__ float ssel[BN][LN];
    __shared__ int   spos[BN];
    const int b    = threadIdx.x >> 5;
    const int lane = threadIdx.x & 31;

    int tok = x[b * LN + lane];
    const float* zw = z_w + (size_t)tok * DMODEL;
    float s = z_b[tok];
    for (int d = 0; d < HGRU; ++d) s += ha[b * HGRU + d] * zw[d];
    for (int d = 0; d < HGRU; ++d) s += hb[b * HGRU + d] * zw[HGRU + d];
    ssel[b][lane] = avail[b * LN + lane] ? s : -3.0e38f;
    __syncthreads();

    if (lane == 0) {
        float best = ssel[b][0]; int pos = 0;
        for (int l = 1; l < LN; ++l)
            if (ssel[b][l] > best) { best = ssel[b][l]; pos = l; }   // first max
        spos[b] = pos;
        avail[b * LN + pos] = 0;
        out[b * LN + idx] = (float)x[b * LN + pos];
    }
    __syncthreads();

    const int pos = spos[b];
    const int tk  = x[b * LN + pos];
    for (int d = lane; d < HIDN; d += 32)
        itinfo[((size_t)b * LP + (idx + 1)) * PADH + d] = emb[(size_t)tk * HIDN + d];
}

// ---------------------------------------------------------------------------
extern "C" void kernel_launch(void* const* d_in, const int* in_sizes, int n_in,
                              void* d_out, int out_size, void* d_ws, size_t ws_size,
                              hipStream_t stream)
{
    const int*   x   = (const int*)d_in[0];
    const float* emb = (const float*)d_in[1];
    GruP e0f{(const float*)d_in[2],  (const float*)d_in[3],  (const float*)d_in[4],  (const float*)d_in[5]};
    GruP e0b{(const float*)d_in[6],  (const float*)d_in[7],  (const float*)d_in[8],  (const float*)d_in[9]};
    GruP e1f{(const float*)d_in[10], (const float*)d_in[11], (const float*)d_in[12], (const float*)d_in[13]};
    GruP e1b{(const float*)d_in[14], (const float*)d_in[15], (const float*)d_in[16], (const float*)d_in[17]};
    GruP dc0{(const float*)d_in[18], (const float*)d_in[19], (const float*)d_in[20], (const float*)d_in[21]};
    GruP dc1{(const float*)d_in[22], (const float*)d_in[23], (const float*)d_in[24], (const float*)d_in[25]};
    const float* conv_w = (const float*)d_in[26];
    const float* conv_b = (const float*)d_in[27];
    const float* o_w = (const float*)d_in[28];  const float* o_b = (const float*)d_in[29];
    const float* q_w = (const float*)d_in[30];  const float* q_b = (const float*)d_in[31];
    const float* k_w = (const float*)d_in[32];  const float* k_b = (const float*)d_in[33];
    const float* v_w = (const float*)d_in[34];  const float* v_b = (const float*)d_in[35];
    const float* z_w = (const float*)d_in[36];  const float* z_b = (const float*)d_in[37];
    float* out = (float*)d_out;

    // workspace carve-up (floats; every segment length is even -> 8B aligned)
    float* ws = (float*)d_ws;
    float* dt     = ws;  ws += LN * BN * PADH;          // [L,B,152]
    float* y0     = ws;  ws += LN * BN * DMODEL;        // [L,B,512]
    float* endt   = ws;  ws += 4 * BN * HGRU;           // [4,B,256]
    float* ha     = ws;  ws += BN * HGRU;
    float* hb     = ws;  ws += BN * HGRU;
    float* itinfo = ws;  ws += BN * LP * PADH;          // [B,33,152]
    float* ko     = ws;  ws += BN * LP * DMODEL;        // [528,512]
    float* qq     = ws;  ws += BN * LP * PADH;
    float* kk     = ws;  ws += BN * LP * PADH;
    float* vv     = ws;  ws += BN * LP * PADH;
    float* data   = ws;  ws += BN * LP * PADH;
    float* w_e0f  = ws;  ws += 768 * PADH;              // repacked [768,152]
    float* w_e0b  = ws;  ws += 768 * PADH;
    float* w_dc0  = ws;  ws += 768 * PADH;
    float* w_o    = ws;  ws += DMODEL * PADH;           // repacked [512,152]
    int*   avail  = (int*)ws;                           // [B,32]

    // ---- pad-repack the K=150 weight matrices (A-side pads are zero, so
    //      B-side pad contents are mathematically inert; zeroed anyway) ----
    repack_kernel<<<64, 256, 0, stream>>>(e0f.wih, w_e0f, 768);
    repack_kernel<<<64, 256, 0, stream>>>(e0b.wih, w_e0b, 768);
    repack_kernel<<<64, 256, 0, stream>>>(dc0.wih, w_dc0, 768);
    repack_kernel<<<64, 256, 0, stream>>>(o_w,     w_o,   DMODEL);

    GruP e0fp{w_e0f, e0f.whh, e0f.bih, e0f.bhh};
    GruP e0bp{w_e0b, e0b.whh, e0b.bih, e0b.bhh};
    GruP dc0p{w_dc0, dc0.whh, dc0.bih, dc0.bhh};

    // ---- encoder ----
    embed_kernel<<<(LN * BN * PADH + 255) / 256, 256, 0, stream>>>(x, emb, dt);
    enc_gru_kernel<PADH><<<2, 256, 0, stream>>>(dt, e0fp, e0bp, y0, endt, 0, 1);
    enc_gru_kernel<DMODEL><<<2, 256, 0, stream>>>(y0, e1f, e1b, nullptr, endt, 2, 3);
    conv_init_kernel<<<32, 256, 0, stream>>>(endt, conv_w, conv_b, ha, hb,
                                             itinfo, data, avail);

    // ---- sequential decode ----
    const int MROW = BN * LP;   // 528
    for (int idx = 0; idx < LN; ++idx) {
        // ko[528,512] = itinfo[528,152] @ o_w^T          (Mt=33, Ng=8 -> 264 waves)
        gemm_wmma4_kernel<PADH><<<33, 256, 0, stream>>>(itinfo, w_o, o_b, ko,
                                                        MROW, DMODEL, DMODEL);
        // q/k/v[528,150(+2)] = ko[528,512] @ {q,k,v}_w^T (Mt=33, Ng=3 -> 99 waves)
        gemm_wmma4_kernel<DMODEL><<<13, 256, 0, stream>>>(ko, q_w, q_b, qq,
                                                          MROW, HIDN, PADH);
        gemm_wmma4_kernel<DMODEL><<<13, 256, 0, stream>>>(ko, k_w, k_b, kk,
                                                          MROW, HIDN, PADH);
        gemm_wmma4_kernel<DMODEL><<<13, 256, 0, stream>>>(ko, v_w, v_b, vv,
                                                          MROW, HIDN, PADH);
        attn_kernel<<<BN, 256, 0, stream>>>(qq, kk, vv, data, idx);
        dec_step_kernel<<<1, 256, 0, stream>>>(data, idx, dc0p, dc1, ha, hb);
        score_kernel<<<1, 512, 0, stream>>>(ha, hb, z_w, z_b, x, emb, avail,
                                            itinfo, out, idx);
    }
}